// LogPolarRepresentation_59914793779876
// MI455X (gfx1250) — compile-verified
//
#include <hip/hip_runtime.h>
#include <math.h>

// Log-polar bilinear resample: img (64,3,512,512) f32 -> out (64,3,360,362) f32.
// Memory-bound gather (~300 MB HBM traffic, ~13us floor @ 23.3 TB/s).
// One thread owns one (theta,r) grid point: computes sampling offsets +
// bilinear weights ONCE in registers, then loops over BCPB planes.
// CDNA5 path: double-buffered GLOBAL_LOAD_ASYNC_TO_LDS_B32 gather pipeline
// (ASYNCcnt-managed, in-order completion), falling back to direct loads +
// global_prefetch_b8 if the async builtins are unavailable.

#define IMG_H   512
#define IMG_W   512
#define RADIUS  362            // int(sqrt(512^2+512^2)/2)
#define NTHETA  360            // 2*REPR_H
#define NBC     192            // 64 * 3
#define BCPB    16             // planes per block (gridDim.z = 12)
#define PLANE   (IMG_H * IMG_W)
#define TPB     192            // 6 wave32s

#if defined(__has_builtin)
#  if __has_builtin(__builtin_amdgcn_global_load_async_to_lds_b32)
#    define HAVE_ASYNC_LDS 1
#  endif
#endif
#ifndef HAVE_ASYNC_LDS
#  define HAVE_ASYNC_LDS 0
#endif

#if HAVE_ASYNC_LDS
typedef __attribute__((address_space(1))) int gas_int;   // global (AS1)
typedef __attribute__((address_space(3))) int las_int;   // LDS    (AS3)

__device__ __forceinline__ void async_gather_b32(const float* gsrc, float* lds_dst) {
    // signature (from hipcc diagnostic): (AS1 int* src, AS3 int* dst, Ii offset, Ii cpol)
    __builtin_amdgcn_global_load_async_to_lds_b32(
        (gas_int*)gsrc, (las_int*)lds_dst, 0, 0);
}

template <int N>
__device__ __forceinline__ void wait_asynccnt() {
#if __has_builtin(__builtin_amdgcn_s_wait_asynccnt)
    __builtin_amdgcn_s_wait_asynccnt(N);
#else
    asm volatile("s_wait_asynccnt %0" :: "n"(N));
#endif
    asm volatile("" ::: "memory");   // keep compiler from hoisting LDS reads
}
#endif

__global__ __launch_bounds__(TPB)
void logpolar_bilinear_kernel(const float* __restrict__ img,
                              float* __restrict__ out) {
#if HAVE_ASYNC_LDS
    // stage[buf][value][lane]: lane-major -> conflict-free ds reads/writes. 6 KB.
    __shared__ float stage[2][4][TPB];
#endif
    const int tid = threadIdx.x;
    const int r   = blockIdx.x * TPB + tid;
    if (r >= RADIUS) return;
    const int theta = blockIdx.y;
    const int bc0   = blockIdx.z * BCPB;

    // ---- grid point (exact f32 replication of the reference math) ----
    const float k   = log2f((float)RADIUS) / (float)RADIUS;   // folded at compile time
    const float rho = exp2f((float)r * k);                    // radius ** (r/radius)
    const float ang = (float)theta * 0.017453292519943295f;   // pi/180 in f32
    const float x   = rho * cosf(ang) + (float)(IMG_W / 2);
    const float y   = rho * sinf(ang) + (float)(IMG_H / 2);

    float gx = x / (float)(IMG_W - 1) * 2.0f - 1.0f;
    float gy = y / (float)(IMG_H - 1) * 2.0f - 1.0f;
    gx = fminf(fmaxf(gx, -1.0f), 1.0f);
    gy = fminf(fmaxf(gy, -1.0f), 1.0f);
    const float ix = (gx + 1.0f) * 0.5f * (float)(IMG_W - 1);
    const float iy = (gy + 1.0f) * 0.5f * (float)(IMG_H - 1);

    const float x0f = floorf(ix);
    const float y0f = floorf(iy);
    const float wx  = ix - x0f;
    const float wy  = iy - y0f;

    int x0i = (int)x0f; x0i = min(max(x0i, 0), IMG_W - 1);
    int y0i = (int)y0f; y0i = min(max(y0i, 0), IMG_H - 1);
    const int x1i = min(x0i + 1, IMG_W - 1);
    const int y1i = min(y0i + 1, IMG_H - 1);

    const int dx    = x1i - x0i;              // 0 or 1
    const int dyo   = (y1i - y0i) * IMG_W;    // 0 or 512
    const int off00 = y0i * IMG_W + x0i;
    const int off01 = off00 + dx;
    const int off10 = off00 + dyo;
    const int off11 = off00 + dyo + dx;

    const float w00 = (1.0f - wx) * (1.0f - wy);
    const float w01 = wx * (1.0f - wy);
    const float w10 = (1.0f - wx) * wy;
    const float w11 = wx * wy;

    const float* __restrict__ p = img + (size_t)bc0 * PLANE;
    float*       __restrict__ o = out + ((size_t)bc0 * NTHETA + theta) * RADIUS + r;

#if HAVE_ASYNC_LDS
    // ---- 2-deep ASYNCcnt-managed gather pipeline (no barriers needed:
    //      each lane touches only its own LDS column) ----
    async_gather_b32(p + off00, &stage[0][0][tid]);
    async_gather_b32(p + off01, &stage[0][1][tid]);
    async_gather_b32(p + off10, &stage[0][2][tid]);
    async_gather_b32(p + off11, &stage[0][3][tid]);

    #pragma unroll
    for (int j = 0; j < BCPB; ++j) {
        const int cur = j & 1;
        const int nxt = cur ^ 1;
        if (j + 1 < BCPB) {
            const float* pn = p + PLANE;
            async_gather_b32(pn + off00, &stage[nxt][0][tid]);
            async_gather_b32(pn + off01, &stage[nxt][1][tid]);
            async_gather_b32(pn + off10, &stage[nxt][2][tid]);
            async_gather_b32(pn + off11, &stage[nxt][3][tid]);
            wait_asynccnt<4>();   // iter j's 4 done (in-order completion)
        } else {
            wait_asynccnt<0>();
        }
        const float v00 = stage[cur][0][tid];
        const float v01 = stage[cur][1][tid];
        const float v10 = stage[cur][2][tid];
        const float v11 = stage[cur][3][tid];
        float res = v00 * w00;
        res = fmaf(v01, w01, res);
        res = fmaf(v10, w10, res);
        res = fmaf(v11, w11, res);
        *o = res;
        p += PLANE;
        o += (size_t)NTHETA * RADIUS;
    }
#else
    // ---- fallback: direct gathers + next-plane global_prefetch_b8 ----
    #pragma unroll
    for (int j = 0; j < BCPB; ++j) {
        if ((j + 1 < BCPB) || (bc0 + BCPB < NBC)) {
            __builtin_prefetch(p + PLANE + off00, 0, 0);
            __builtin_prefetch(p + PLANE + off10, 0, 0);
        }
        const float v00 = p[off00];
        const float v01 = p[off01];
        const float v10 = p[off10];
        const float v11 = p[off11];
        float res = v00 * w00;
        res = fmaf(v01, w01, res);
        res = fmaf(v10, w10, res);
        res = fmaf(v11, w11, res);
        *o = res;
        p += PLANE;
        o += (size_t)NTHETA * RADIUS;
    }
#endif
}

extern "C" void kernel_launch(void* const* d_in, const int* in_sizes, int n_in,
                              void* d_out, int out_size, void* d_ws, size_t ws_size,
                              hipStream_t stream) {
    (void)in_sizes; (void)n_in; (void)d_ws; (void)ws_size; (void)out_size;
    const float* img = (const float*)d_in[0];
    float*       out = (float*)d_out;

    dim3 block(TPB, 1, 1);
    dim3 grid((RADIUS + TPB - 1) / TPB,   // 2 blocks cover r = 0..361
              NTHETA,                     // 360 theta rows
              NBC / BCPB);                // 12 groups of 16 planes
    logpolar_bilinear_kernel<<<grid, block, 0, stream>>>(img, out);
}